// MultiTypeAttentionDist_87789131530407
// MI455X (gfx1250) — compile-verified
//
#include <hip/hip_runtime.h>

#define BATCH 4096
#define NODES 256
#define FDIM  128
#define HID   16

#define XS_STRIDE 132   // 128 + 4 pad -> 4-bank stride per row, conflict-free A loads
#define A1_STRIDE 260   // 256 + 4 pad, same trick for transposed B operand

typedef __attribute__((ext_vector_type(2))) float v2f;
typedef __attribute__((ext_vector_type(4))) float v4f;
typedef __attribute__((ext_vector_type(8))) float v8f;

__device__ __forceinline__ float lrelu(float v) { return v > 0.0f ? v : 0.01f * v; }

__global__ __launch_bounds__(256) void gat_attn_kernel(
    const float* __restrict__ x,          // [B, N, F]
    const float* __restrict__ a1,         // [T, 2F, H]
    const float* __restrict__ a2,         // [T, H, 1]
    const float* __restrict__ adj,        // [N]
    const int*   __restrict__ node_index, // [1]
    const int*   __restrict__ type_index, // [B]
    float*       __restrict__ out)        // [B, N]
{
    extern __shared__ float smem[];
    float* xs   = smem;                       // [NODES][XS_STRIDE]
    float* a1t  = xs  + NODES * XS_STRIDE;    // [HID][A1_STRIDE]  (a1[t] transposed)
    float* a2s  = a1t + HID * A1_STRIDE;      // [HID]
    float* base = a2s + HID;                  // [HID]
    float* es   = base + HID;                 // [NODES]
    float* red  = es  + NODES;                // [8]

    const int b   = blockIdx.x;
    const int tid = threadIdx.x;
    const int t   = type_index[b];

    // ---- stage x[b,:,:] into LDS, coalesced 16B loads ----
    const v4f* xg = (const v4f*)(x + (size_t)b * NODES * FDIM);
    for (int i = tid; i < NODES * FDIM / 4; i += 256) {
        v4f v = xg[i];
        const int row = i >> 5;            // FDIM/4 = 32 float4 per row
        const int col = (i & 31) << 2;
        float* d = xs + row * XS_STRIDE + col;
        d[0] = v.x; d[1] = v.y; d[2] = v.z; d[3] = v.w;
    }
    // ---- stage a1[t] transposed: a1t[h][c] = a1[t][c][h] ----
    const float* a1g = a1 + (size_t)t * (2 * FDIM) * HID;
    for (int i = tid; i < 2 * FDIM * HID; i += 256) {
        const int c = i >> 4, h = i & 15;
        a1t[h * A1_STRIDE + c] = a1g[c * HID + h];
    }
    if (tid < HID) a2s[tid] = a2[t * HID + tid];
    __syncthreads();

    // ---- base[h] = sum_{c<F} x[b, node_index, c] * a1[t, c, h] (shared by all rows) ----
    if (tid < HID) {
        const float* xrow = xs + node_index[0] * XS_STRIDE;
        const float* arow = a1t + tid * A1_STRIDE;
        float s = 0.0f;
        #pragma unroll 8
        for (int c = 0; c < FDIM; ++c) s += xrow[c] * arow[c];
        base[tid] = s;
    }
    __syncthreads();

    // ---- per-wave GEMM: two 16-node tiles, K=128 via v_wmma_f32_16x16x4_f32 ----
    const int wave = tid >> 5;
    const int lane = tid & 31;
    const int l16  = lane & 15;
    const int hi   = lane >> 4;               // selects K sub-pair {0,1} vs {2,3}

    const float bc = base[l16];               // C layout: N = lane%16 for all 8 VGPRs
    v8f acc0, acc1;
    #pragma unroll
    for (int j = 0; j < 8; ++j) { acc0[j] = bc; acc1[j] = bc; }

    // A: row M = lane%16 (tile base + l16), VGPR0/1 = K = 2*hi, 2*hi+1
    const float* aptr0 = xs + (wave * 32 + l16) * XS_STRIDE + 2 * hi;
    const float* aptr1 = aptr0 + 16 * XS_STRIDE;
    // B: N = lane%16, K rows 2*hi, 2*hi+1 of a1[t, 128+k+..., :]
    const float* bptr  = a1t + l16 * A1_STRIDE + FDIM + 2 * hi;

    #pragma unroll 4
    for (int k = 0; k < FDIM; k += 4) {
        v2f av0 = *(const v2f*)(aptr0 + k);
        v2f av1 = *(const v2f*)(aptr1 + k);
        v2f bv  = *(const v2f*)(bptr  + k);
        acc0 = __builtin_amdgcn_wmma_f32_16x16x4_f32(false, av0, false, bv,
                                                     (short)0, acc0, false, false);
        acc1 = __builtin_amdgcn_wmma_f32_16x16x4_f32(false, av1, false, bv,
                                                     (short)0, acc1, false, false);
    }

    // ---- e[n] = lrelu( sum_h lrelu(hvals) * a2[t,h] ), half-wave reductions ----
    const float a2v = a2s[l16];
    const int nb = wave * 32 + hi * 8;        // VGPR j <-> node nb + j (tile0)
    #pragma unroll
    for (int j = 0; j < 8; ++j) {
        float p = lrelu(acc0[j]) * a2v;
        p += __shfl_xor(p, 1, 32);
        p += __shfl_xor(p, 2, 32);
        p += __shfl_xor(p, 4, 32);
        p += __shfl_xor(p, 8, 32);
        if (l16 == 0) es[nb + j] = lrelu(p);
        float q = lrelu(acc1[j]) * a2v;
        q += __shfl_xor(q, 1, 32);
        q += __shfl_xor(q, 2, 32);
        q += __shfl_xor(q, 4, 32);
        q += __shfl_xor(q, 8, 32);
        if (l16 == 0) es[nb + 16 + j] = lrelu(q);
    }
    __syncthreads();

    // ---- masked softmax over N=256, one thread per node ----
    const int n = tid;
    const float adjv = adj[n];
    const bool  on   = adjv > 0.0f;
    float ev = on ? es[n] : -3.0e38f;

    float m = ev;
    m = fmaxf(m, __shfl_xor(m, 1,  32));
    m = fmaxf(m, __shfl_xor(m, 2,  32));
    m = fmaxf(m, __shfl_xor(m, 4,  32));
    m = fmaxf(m, __shfl_xor(m, 8,  32));
    m = fmaxf(m, __shfl_xor(m, 16, 32));
    if (lane == 0) red[wave] = m;
    __syncthreads();
    float mx = red[0];
    #pragma unroll
    for (int i = 1; i < 8; ++i) mx = fmaxf(mx, red[i]);
    __syncthreads();

    const float p = on ? __expf(ev - mx) : 0.0f;
    float s = p;
    s += __shfl_xor(s, 1,  32);
    s += __shfl_xor(s, 2,  32);
    s += __shfl_xor(s, 4,  32);
    s += __shfl_xor(s, 8,  32);
    s += __shfl_xor(s, 16, 32);
    if (lane == 0) red[wave] = s;
    __syncthreads();
    float tot = red[0];
    #pragma unroll
    for (int i = 1; i < 8; ++i) tot += red[i];

    out[(size_t)b * NODES + n] = p / tot;
}

extern "C" void kernel_launch(void* const* d_in, const int* in_sizes, int n_in,
                              void* d_out, int out_size, void* d_ws, size_t ws_size,
                              hipStream_t stream) {
    (void)in_sizes; (void)n_in; (void)out_size; (void)d_ws; (void)ws_size;
    const float* x          = (const float*)d_in[0];
    const float* a1         = (const float*)d_in[1];
    const float* a2         = (const float*)d_in[2];
    const float* adj        = (const float*)d_in[3];
    const int*   node_index = (const int*)d_in[4];
    const int*   type_index = (const int*)d_in[5];
    float*       out        = (float*)d_out;

    const size_t shmem = (size_t)(NODES * XS_STRIDE + HID * A1_STRIDE +
                                  HID + HID + NODES + 8) * sizeof(float);
    gat_attn_kernel<<<BATCH, 256, shmem, stream>>>(x, a1, a2, adj,
                                                   node_index, type_index, out);
}